// modmax_loss_44178033607235
// MI455X (gfx1250) — compile-verified
//
#include <hip/hip_runtime.h>

#define N_NODES 8192
#define C_CLS   16

typedef __attribute__((ext_vector_type(2))) float v2f;
typedef __attribute__((ext_vector_type(8))) float v8f;

// Workspace layout (floats):
//   ws[0]        quad accumulator (f32 atomics)
//   ws[1]        s accumulator    (int atomics, reinterpreted)
//   ws[2..17]    dH[16]           (f32 atomics)
//   ws[64..]     Ht[16][8192]     transposed H (column-major H)
#define WS_QUAD 0
#define WS_S    1
#define WS_DH   2
#define WS_HT   64

// ---------------------------------------------------------------------------
// Prologue: zero the accumulators and build Ht[c][j] = H[j][c].
// ---------------------------------------------------------------------------
__global__ void modmax_prep(const float* __restrict__ H, float* __restrict__ ws) {
    const int idx = blockIdx.x * 256 + threadIdx.x;      // 0 .. 131071
    if (idx < 64) ws[idx] = 0.0f;                        // zero accumulators
    const int c = idx >> 13;                             // 0..15
    const int j = idx & (N_NODES - 1);                   // 0..8191
    ws[WS_HT + c * N_NODES + j] = H[(size_t)j * C_CLS + c];
}

// ---------------------------------------------------------------------------
// Main kernel. Each wave owns one 16-row band of Y = adj @ H and a chunk of
// 32 j-tiles (512 columns). Per 16x16 tile:
//   - adj mask built from 4 NT b64 loads (pairs land in A-operand K slots)
//   - B operand (H rows j0..j0+15 striped over K) from Ht with 4 b64 loads
//   - 4 chained v_wmma_f32_16x16x4_f32 accumulate Y across the whole chunk
//   - s counted on the scalar pipe via ballot+popcount (co-issues with VALU)
// Epilogue: quad += <Hi, Y_band>, dH[c] += column sums of Y_band, 3 atomics.
//
// f32 WMMA 16x16x4 operand striping (ISA 7.12.2):
//   A (16x4): lanes 0-15 -> M=lane, VGPR0=K+0, VGPR1=K+1; lanes 16-31 -> K+2,K+3
//   B (4x16): lanes 0-15 -> N=lane, VGPR0=K+0, VGPR1=K+1; lanes 16-31 -> K+2,K+3
//   C/D     : VGPR k, lanes 0-15 -> (M=k, N=lane); lanes 16-31 -> (M=k+8, N=lane-16)
// ---------------------------------------------------------------------------
__global__ __launch_bounds__(256) void modmax_main(const float* __restrict__ H,
                                                   const float* __restrict__ A,
                                                   float* __restrict__ ws) {
    const float* Ht = ws + WS_HT;

    const int lane = threadIdx.x & 31;
    const int wid  = threadIdx.x >> 5;
    const int l15  = lane & 15;
    const int koff = (lane >> 4) << 1;                  // 0 (lanes 0-15) or 2

    const int i0 = (blockIdx.x * 8 + wid) * 16;         // this wave's row band
    const int j0 = blockIdx.y * 32 * 16;                // first column of chunk

    // Per-lane streaming pointers; both advance 16 floats per j-tile.
    const float* ap = A  + (size_t)(i0 + l15) * N_NODES + (size_t)(j0 + koff);
    const float* hp = Ht + (size_t)l15        * N_NODES + (size_t)(j0 + koff);

    v8f yacc = {};                                      // Y band accumulator
    int scount = 0;                                     // uniform popcount of adj

#pragma unroll 4
    for (int t = 0; t < 32; ++t) {
        v2f b0 = *(const v2f*)(hp + 0);
        v2f b1 = *(const v2f*)(hp + 4);
        v2f b2 = *(const v2f*)(hp + 8);
        v2f b3 = *(const v2f*)(hp + 12);

        v2f r0 = __builtin_nontemporal_load((const v2f*)(ap + 0));
        v2f r1 = __builtin_nontemporal_load((const v2f*)(ap + 4));
        v2f r2 = __builtin_nontemporal_load((const v2f*)(ap + 8));
        v2f r3 = __builtin_nontemporal_load((const v2f*)(ap + 12));

        // Build {1.0, 0.0} mask (v_cmp + v_cndmask) and count positives on
        // the scalar pipe (ballot reuses the same v_cmp; popcount is SALU).
        v2f m0, m1, m2, m3;
        {
            const bool e0 = r0.x > 0.0f, e1 = r0.y > 0.0f;
            const bool e2 = r1.x > 0.0f, e3 = r1.y > 0.0f;
            const bool e4 = r2.x > 0.0f, e5 = r2.y > 0.0f;
            const bool e6 = r3.x > 0.0f, e7 = r3.y > 0.0f;
            m0.x = e0 ? 1.0f : 0.0f;  m0.y = e1 ? 1.0f : 0.0f;
            m1.x = e2 ? 1.0f : 0.0f;  m1.y = e3 ? 1.0f : 0.0f;
            m2.x = e4 ? 1.0f : 0.0f;  m2.y = e5 ? 1.0f : 0.0f;
            m3.x = e6 ? 1.0f : 0.0f;  m3.y = e7 ? 1.0f : 0.0f;
            scount += (int)__popcll(__ballot(e0)) + (int)__popcll(__ballot(e1))
                    + (int)__popcll(__ballot(e2)) + (int)__popcll(__ballot(e3))
                    + (int)__popcll(__ballot(e4)) + (int)__popcll(__ballot(e5))
                    + (int)__popcll(__ballot(e6)) + (int)__popcll(__ballot(e7));
        }

        // Y += adj_tile @ H_rows(j-tile):  K=16 via 4 chained 16x16x4 WMMAs.
        yacc = __builtin_amdgcn_wmma_f32_16x16x4_f32(false, m0, false, b0, (short)0, yacc, false, false);
        yacc = __builtin_amdgcn_wmma_f32_16x16x4_f32(false, m1, false, b1, (short)0, yacc, false, false);
        yacc = __builtin_amdgcn_wmma_f32_16x16x4_f32(false, m2, false, b2, (short)0, yacc, false, false);
        yacc = __builtin_amdgcn_wmma_f32_16x16x4_f32(false, m3, false, b3, (short)0, yacc, false, false);

        ap += 16;
        hp += 16;
    }

    // ---- epilogue: fold Y band into quad / dH, s into the int slot --------

    // dH[c] partial: sum Y over the band's 16 rows (8 per lane-half).
    float dh = yacc[0] + yacc[1] + yacc[2] + yacc[3]
             + yacc[4] + yacc[5] + yacc[6] + yacc[7];
    dh += __shfl_xor(dh, 16, 32);
    if (lane < 16) atomicAdd(&ws[WS_DH + lane], dh);

    // quad partial: elementwise dot of Y band with H rows of the band.
    const int mbase = i0 + ((lane >> 4) << 3);
    float q = 0.0f;
#pragma unroll
    for (int k = 0; k < 8; ++k)
        q += H[(size_t)(mbase + k) * C_CLS + l15] * yacc[k];
    for (int off = 16; off > 0; off >>= 1)
        q += __shfl_xor(q, off, 32);

    if (lane == 0) {
        atomicAdd(&ws[WS_QUAD], q);
        atomicAdd((int*)ws + WS_S, scount);   // exact integer count
    }
}

// ---------------------------------------------------------------------------
// Final: loss = (quad - dot(dH,dH)/s) / s   (f64 scalar math, one thread).
// ---------------------------------------------------------------------------
__global__ void modmax_final(const float* __restrict__ ws, float* __restrict__ out) {
    const double quad = (double)ws[WS_QUAD];
    const double s    = (double)(((const int*)ws)[WS_S]);
    double dot = 0.0;
#pragma unroll
    for (int c = 0; c < C_CLS; ++c) {
        const double v = (double)ws[WS_DH + c];
        dot += v * v;
    }
    out[0] = (float)((quad - dot / s) / s);
}

// ---------------------------------------------------------------------------
extern "C" void kernel_launch(void* const* d_in, const int* in_sizes, int n_in,
                              void* d_out, int out_size, void* d_ws, size_t ws_size,
                              hipStream_t stream) {
    const float* H = (const float*)d_in[0];   // H_i: [8192, 16] f32
    const float* A = (const float*)d_in[1];   // A_i: [8192, 8192] f32
    float* ws  = (float*)d_ws;                // needs 64 + 16*8192 floats (~513 KB)
    float* out = (float*)d_out;

    modmax_prep<<<(C_CLS * N_NODES) / 256, 256, 0, stream>>>(H, ws);

    dim3 grid(64, 16);                        // 64 band-groups x 16 j-chunks
    modmax_main<<<grid, dim3(256), 0, stream>>>(H, A, ws);

    modmax_final<<<1, 1, 0, stream>>>(ws, out);
}